// GAT_34342558498882
// MI455X (gfx1250) — compile-verified
//
#include <hip/hip_runtime.h>
#include <hip/hip_bf16.h>
#include <math.h>

// ---------------------------------------------------------------------------
// 3-layer single-head GATConv (PyG semantics: add_self_loops, leaky_relu 0.2,
// segment softmax over incoming edges) + linear classifier.
// N=100000 nodes, E=6.4M edges, dims 128 -> 4 -> 4 -> 2 -> 8.
// Edge passes dominate; indices are compacted to int32 pairs (L2-resident).
// Layer-1 GEMM uses V_WMMA_F32_16X16X4_F32 with a branchless inner loop.
// ---------------------------------------------------------------------------

#define NEG_SLOPE 0.2f

typedef __attribute__((ext_vector_type(2))) float v2f;
typedef __attribute__((ext_vector_type(8))) float v8f;

__device__ __forceinline__ float lrelu(float v) {
    return v >= 0.0f ? v : NEG_SLOPE * v;
}

// Sign-split float atomic max (monotone in both domains; init must be -inf).
__device__ __forceinline__ void atomicMaxFloat(float* addr, float val) {
    if (!(val < 0.0f)) {
        atomicMax((int*)addr, __float_as_int(val));
    } else {
        atomicMin((unsigned int*)addr, __float_as_uint(val));
    }
}

// ---------------------------------------------------------------------------
// Layer-1 GEMM: h = x[N,128] @ W1[128,4]  via V_WMMA_F32_16X16X4_F32.
// One wave per 16-row tile; K=128 in 32 steps. Addresses are CLAMPED, not
// predicated: garbage A-rows >= N / B-cols >= 4 only feed C entries that are
// never stored, so the inner loop is branchless (no EXEC save/restore).
// ISA VGPR layouts (CDNA5 ISA §7.12.2):
//   A 16x4:  lane(m=lane&15, kh=lane>>4) holds A[m][2kh], A[m][2kh+1]
//   B 4x16:  lane(n=lane&15, kh)         holds B[2kh][n], B[2kh+1][n]
//   C 16x16: vgpr j of lane holds (M=j+8*kh, N=lane&15)
// ---------------------------------------------------------------------------
__global__ void gat_gemm1_wmma(const float* __restrict__ x,
                               const float* __restrict__ W,   // [128,4] row-major
                               float* __restrict__ hbuf,      // [N,4]
                               int n_nodes) {
    const int wave = threadIdx.x >> 5;
    const int lane = threadIdx.x & 31;
    const long long tile = (long long)blockIdx.x * (blockDim.x >> 5) + wave;
    const long long base = tile * 16;
    if (base >= n_nodes) return;                 // wave-uniform exit

    const int m  = lane & 15;                    // row (A) / col (B,C)
    const int kh = lane >> 4;                    // K-half selector

    long long row = base + m;
    if (row >= n_nodes) row = n_nodes - 1;       // clamp (never stored)
    const float* xrow = x + row * 128 + 2 * kh;

    const int col = m & 3;                       // clamp (cols 4..15 never stored)
    const float* wcol = W + 2 * kh * 4 + col;    // &W[2kh][col]

    v8f c = {0.f, 0.f, 0.f, 0.f, 0.f, 0.f, 0.f, 0.f};

    #pragma unroll
    for (int k0 = 0; k0 < 128; k0 += 4) {
        v2f a, b;
        a.x = xrow[k0];                          // A[m][k0+2kh]
        a.y = xrow[k0 + 1];                      // A[m][k0+2kh+1]
        b.x = wcol[k0 * 4];                      // W[k0+2kh][col]
        b.y = wcol[k0 * 4 + 4];                  // W[k0+2kh+1][col]
        c = __builtin_amdgcn_wmma_f32_16x16x4_f32(
                false, a, false, b, (short)0, c, false, false);
    }

    if (m < 4) {
        #pragma unroll
        for (int j = 0; j < 8; ++j) {
            long long r = base + j + 8 * kh;
            if (r < (long long)n_nodes) hbuf[r * 4 + m] = c[j];
        }
    }
}

// Scalar transform for tiny layers (4x4, 4x2): hbuf = xin @ W
template <int FIN, int FOUT>
__global__ void node_transform(const float* __restrict__ xin,
                               const float* __restrict__ W,   // [FIN,FOUT]
                               float* __restrict__ hbuf, int n) {
    int i = blockIdx.x * blockDim.x + threadIdx.x;
    if (i >= n) return;
    float xl[FIN];
    #pragma unroll
    for (int f = 0; f < FIN; ++f) xl[f] = xin[(long long)i * FIN + f];
    #pragma unroll
    for (int o = 0; o < FOUT; ++o) {
        float acc = 0.f;
        #pragma unroll
        for (int f = 0; f < FIN; ++f) acc += xl[f] * W[f * FOUT + o];
        hbuf[(long long)i * FOUT + o] = acc;
    }
}

// Per-node attention coefficients + per-layer state init.
template <int FOUT>
__global__ void node_alpha_init(const float* __restrict__ hbuf,
                                const float* __restrict__ a_src,
                                const float* __restrict__ a_dst,
                                float* __restrict__ alpha_s,
                                float* __restrict__ alpha_d,
                                float* __restrict__ e_max,
                                float* __restrict__ denom,
                                float* __restrict__ agg, int n) {
    int i = blockIdx.x * blockDim.x + threadIdx.x;
    if (i >= n) return;
    float s = 0.f, d = 0.f;
    #pragma unroll
    for (int f = 0; f < FOUT; ++f) {
        float hv = hbuf[(long long)i * FOUT + f];
        s += hv * a_src[f];
        d += hv * a_dst[f];
    }
    alpha_s[i] = s;
    alpha_d[i] = d;
    e_max[i]   = __int_as_float(0xff800000u);    // -inf
    denom[i]   = 0.f;
    #pragma unroll
    for (int f = 0; f < FOUT; ++f) agg[(long long)i * FOUT + f] = 0.f;
}

// ---- int64 -> packed int32 (src,dst) pairs, self-loops appended -----------
__global__ void convert_edges(const long long* __restrict__ src,
                              const long long* __restrict__ dst,
                              int2* __restrict__ pairs,
                              long long E, long long ET) {
    long long e = (long long)blockIdx.x * blockDim.x + threadIdx.x;
    if (e >= ET) return;
    int2 p;
    if (e < E) { p.x = (int)src[e]; p.y = (int)dst[e]; }
    else       { p.x = p.y = (int)(e - E); }
    pairs[e] = p;
}

// ---- edge passes, int32-pair path (L2-resident index stream) --------------
__global__ void edge_max32(const int2* __restrict__ pairs,
                           const float* __restrict__ alpha_s,
                           const float* __restrict__ alpha_d,
                           float* __restrict__ e_max, long long ET) {
    long long e = (long long)blockIdx.x * blockDim.x + threadIdx.x;
    if (e >= ET) return;
    int2 p = pairs[e];
    atomicMaxFloat(&e_max[p.y], lrelu(alpha_s[p.x] + alpha_d[p.y]));
}

__global__ void edge_expsum32(const int2* __restrict__ pairs,
                              const float* __restrict__ alpha_s,
                              const float* __restrict__ alpha_d,
                              const float* __restrict__ e_max,
                              float* __restrict__ denom, long long ET) {
    long long e = (long long)blockIdx.x * blockDim.x + threadIdx.x;
    if (e >= ET) return;
    int2 p = pairs[e];
    float v = lrelu(alpha_s[p.x] + alpha_d[p.y]);
    atomicAdd(&denom[p.y], expf(v - e_max[p.y]));
}

template <int FOUT>
__global__ void edge_aggregate32(const int2* __restrict__ pairs,
                                 const float* __restrict__ alpha_s,
                                 const float* __restrict__ alpha_d,
                                 const float* __restrict__ e_max,
                                 const float* __restrict__ denom,
                                 const float* __restrict__ hbuf,
                                 float* __restrict__ agg, long long ET) {
    long long e = (long long)blockIdx.x * blockDim.x + threadIdx.x;
    if (e >= ET) return;
    int2 p = pairs[e];
    float v = lrelu(alpha_s[p.x] + alpha_d[p.y]);
    float alpha = expf(v - e_max[p.y]) / denom[p.y];
    #pragma unroll
    for (int f = 0; f < FOUT; ++f)
        atomicAdd(&agg[(long long)p.y * FOUT + f],
                  hbuf[(long long)p.x * FOUT + f] * alpha);
}

// ---- edge passes, int64 fallback (if workspace too small for pairs) -------
__device__ __forceinline__ void edge_ids(long long e, const long long* src,
                                         const long long* dst, long long E,
                                         long long& s, long long& d) {
    if (e < E) { s = src[e]; d = dst[e]; }
    else       { s = e - E;  d = e - E;  }
}

__global__ void edge_max64(const long long* __restrict__ src,
                           const long long* __restrict__ dst,
                           const float* __restrict__ alpha_s,
                           const float* __restrict__ alpha_d,
                           float* __restrict__ e_max, long long E, long long ET) {
    long long e = (long long)blockIdx.x * blockDim.x + threadIdx.x;
    if (e >= ET) return;
    long long s, d; edge_ids(e, src, dst, E, s, d);
    atomicMaxFloat(&e_max[d], lrelu(alpha_s[s] + alpha_d[d]));
}

__global__ void edge_expsum64(const long long* __restrict__ src,
                              const long long* __restrict__ dst,
                              const float* __restrict__ alpha_s,
                              const float* __restrict__ alpha_d,
                              const float* __restrict__ e_max,
                              float* __restrict__ denom, long long E, long long ET) {
    long long e = (long long)blockIdx.x * blockDim.x + threadIdx.x;
    if (e >= ET) return;
    long long s, d; edge_ids(e, src, dst, E, s, d);
    float v = lrelu(alpha_s[s] + alpha_d[d]);
    atomicAdd(&denom[d], expf(v - e_max[d]));
}

template <int FOUT>
__global__ void edge_aggregate64(const long long* __restrict__ src,
                                 const long long* __restrict__ dst,
                                 const float* __restrict__ alpha_s,
                                 const float* __restrict__ alpha_d,
                                 const float* __restrict__ e_max,
                                 const float* __restrict__ denom,
                                 const float* __restrict__ hbuf,
                                 float* __restrict__ agg, long long E, long long ET) {
    long long e = (long long)blockIdx.x * blockDim.x + threadIdx.x;
    if (e >= ET) return;
    long long s, d; edge_ids(e, src, dst, E, s, d);
    float v = lrelu(alpha_s[s] + alpha_d[d]);
    float alpha = expf(v - e_max[d]) / denom[d];
    #pragma unroll
    for (int f = 0; f < FOUT; ++f)
        atomicAdd(&agg[d * FOUT + f], hbuf[s * FOUT + f] * alpha);
}

template <int FOUT>
__global__ void finalize_tanh(const float* __restrict__ agg,
                              const float* __restrict__ b,
                              float* __restrict__ out, int n) {
    int i = blockIdx.x * blockDim.x + threadIdx.x;
    if (i >= n) return;
    #pragma unroll
    for (int f = 0; f < FOUT; ++f)
        out[(long long)i * FOUT + f] = tanhf(agg[(long long)i * FOUT + f] + b[f]);
}

__global__ void classifier(const float* __restrict__ h3,
                           const float* __restrict__ Wc,  // [2,8]
                           const float* __restrict__ bc,
                           float* __restrict__ out, int n) {
    int i = blockIdx.x * blockDim.x + threadIdx.x;
    if (i >= n) return;
    float a = h3[(long long)i * 2];
    float b = h3[(long long)i * 2 + 1];
    #pragma unroll
    for (int c = 0; c < 8; ++c)
        out[(long long)i * 8 + c] = a * Wc[c] + b * Wc[8 + c] + bc[c];
}

extern "C" void kernel_launch(void* const* d_in, const int* in_sizes, int n_in,
                              void* d_out, int out_size, void* d_ws, size_t ws_size,
                              hipStream_t stream) {
    const int       N  = in_sizes[0] / 128;           // 100000
    const long long E  = in_sizes[1] / 2;             // 6400000
    const long long ET = E + N;                       // + self loops

    const float*     x   = (const float*)d_in[0];
    const long long* ei  = (const long long*)d_in[1];
    const long long* src = ei;
    const long long* dst = ei + E;
    const float *W1 = (const float*)d_in[2],  *a1s = (const float*)d_in[3],
                *a1d = (const float*)d_in[4], *b1  = (const float*)d_in[5];
    const float *W2 = (const float*)d_in[6],  *a2s = (const float*)d_in[7],
                *a2d = (const float*)d_in[8], *b2  = (const float*)d_in[9];
    const float *W3 = (const float*)d_in[10], *a3s = (const float*)d_in[11],
                *a3d = (const float*)d_in[12], *b3 = (const float*)d_in[13];
    const float *Wc = (const float*)d_in[14], *bc  = (const float*)d_in[15];

    float* out    = (float*)d_out;
    float* h1     = out;                               // [N,4]
    float* h2     = out + (size_t)N * 4;               // [N,4]
    float* h3     = out + (size_t)N * 8;               // [N,2]
    float* logits = out + (size_t)N * 10;              // [N,8]

    float* ws      = (float*)d_ws;
    float* alpha_s = ws;
    float* alpha_d = ws + (size_t)N;
    float* e_max   = ws + (size_t)2 * N;
    float* denom   = ws + (size_t)3 * N;
    float* agg     = ws + (size_t)4 * N;               // [N,4] max
    float* hbuf    = ws + (size_t)8 * N;               // [N,4] max
    int2*  pairs   = (int2*)(ws + (size_t)12 * N);     // [ET] packed (src,dst)

    const size_t need_pairs = (size_t)12 * N * sizeof(float) + (size_t)ET * sizeof(int2);
    const bool use32 = (ws_size >= need_pairs);

    const int TB = 256;
    const int nodeBlocks = (N + TB - 1) / TB;
    const int edgeBlocks = (int)((ET + TB - 1) / TB);
    const int tiles      = (N + 15) / 16;
    const int gemmBlocks = (tiles + 3) / 4;            // 4 waves / block

    if (use32)
        convert_edges<<<edgeBlocks, TB, 0, stream>>>(src, dst, pairs, E, ET);

    struct Layer {
        const float *W, *as, *ad, *b; float* outp; int fin, fout;
    };
    const Layer L[3] = {
        { W1, a1s, a1d, b1, h1, 128, 4 },
        { W2, a2s, a2d, b2, h2, 4,   4 },
        { W3, a3s, a3d, b3, h3, 4,   2 },
    };
    const float* lin[3] = { x, h1, h2 };

    for (int l = 0; l < 3; ++l) {
        // transform: h = x_in @ W
        if (l == 0)
            gat_gemm1_wmma<<<gemmBlocks, 128, 0, stream>>>(x, L[0].W, hbuf, N);
        else if (L[l].fout == 4)
            node_transform<4, 4><<<nodeBlocks, TB, 0, stream>>>(lin[l], L[l].W, hbuf, N);
        else
            node_transform<4, 2><<<nodeBlocks, TB, 0, stream>>>(lin[l], L[l].W, hbuf, N);

        // attention coefficients + state init
        if (L[l].fout == 4)
            node_alpha_init<4><<<nodeBlocks, TB, 0, stream>>>(hbuf, L[l].as, L[l].ad,
                alpha_s, alpha_d, e_max, denom, agg, N);
        else
            node_alpha_init<2><<<nodeBlocks, TB, 0, stream>>>(hbuf, L[l].as, L[l].ad,
                alpha_s, alpha_d, e_max, denom, agg, N);

        // segment softmax + aggregation
        if (use32) {
            edge_max32<<<edgeBlocks, TB, 0, stream>>>(pairs, alpha_s, alpha_d, e_max, ET);
            edge_expsum32<<<edgeBlocks, TB, 0, stream>>>(pairs, alpha_s, alpha_d,
                                                         e_max, denom, ET);
            if (L[l].fout == 4)
                edge_aggregate32<4><<<edgeBlocks, TB, 0, stream>>>(pairs, alpha_s, alpha_d,
                    e_max, denom, hbuf, agg, ET);
            else
                edge_aggregate32<2><<<edgeBlocks, TB, 0, stream>>>(pairs, alpha_s, alpha_d,
                    e_max, denom, hbuf, agg, ET);
        } else {
            edge_max64<<<edgeBlocks, TB, 0, stream>>>(src, dst, alpha_s, alpha_d,
                                                      e_max, E, ET);
            edge_expsum64<<<edgeBlocks, TB, 0, stream>>>(src, dst, alpha_s, alpha_d,
                                                         e_max, denom, E, ET);
            if (L[l].fout == 4)
                edge_aggregate64<4><<<edgeBlocks, TB, 0, stream>>>(src, dst, alpha_s,
                    alpha_d, e_max, denom, hbuf, agg, E, ET);
            else
                edge_aggregate64<2><<<edgeBlocks, TB, 0, stream>>>(src, dst, alpha_s,
                    alpha_d, e_max, denom, hbuf, agg, E, ET);
        }

        // bias + tanh -> layer output (lives in d_out)
        if (L[l].fout == 4)
            finalize_tanh<4><<<nodeBlocks, TB, 0, stream>>>(agg, L[l].b, L[l].outp, N);
        else
            finalize_tanh<2><<<nodeBlocks, TB, 0, stream>>>(agg, L[l].b, L[l].outp, N);
    }

    classifier<<<nodeBlocks, TB, 0, stream>>>(h3, Wc, bc, logits, N);
}